// GraphClassifier_29557964931462
// MI455X (gfx1250) — compile-verified
//
#include <hip/hip_runtime.h>

#define B_GRAPHS 1024
#define F_FRAMES 25
#define NPF_K    10
#define P_NODES  250      // F * NPF
#define D_IN     64
#define H_DIM    128
#define C_CLS    10
#define NBINS    256

typedef float v2f __attribute__((ext_vector_type(2)));
typedef float v8f __attribute__((ext_vector_type(8)));

// ---------------------------------------------------------------------------
// 1) Per-graph lower-median frame + selected-node count (histogram select)
// ---------------------------------------------------------------------------
__global__ __launch_bounds__(256) void median_kernel(const int* __restrict__ frame_mask,
                                                     int* __restrict__ med,
                                                     int* __restrict__ den) {
    __shared__ int hist[NBINS];
    const int g = blockIdx.x;
    for (int i = threadIdx.x; i < NBINS; i += 256) hist[i] = 0;
    __syncthreads();
    const int base = g * P_NODES;
    for (int i = threadIdx.x; i < P_NODES; i += 256) {
        int v = frame_mask[base + i] & (NBINS - 1);
        atomicAdd(&hist[v], 1);
    }
    __syncthreads();
    if (threadIdx.x == 0) {
        const int target = (P_NODES + 1) / 2;   // lower median, 1-based rank
        int cum = 0, m = 0, d = 1;
        for (int v = 0; v < NBINS; ++v) {
            cum += hist[v];
            if (cum >= target) { m = v; d = hist[v]; break; }
        }
        med[g] = m;
        den[g] = d;
    }
}

// ---------------------------------------------------------------------------
// 2) Zero counter + pooled accumulator (ws is poisoned, must re-init per call)
// ---------------------------------------------------------------------------
__global__ __launch_bounds__(256) void zero_kernel(float* __restrict__ num,
                                                   int* __restrict__ counter, int n) {
    int i = blockIdx.x * 256 + threadIdx.x;
    if (i == 0) *counter = 0;
    if (i < n) num[i] = 0.0f;
}

// ---------------------------------------------------------------------------
// 3) Build compact list of median-frame nodes; seed agg row with x[node]
//    (implements the "x + agg" self term)
// ---------------------------------------------------------------------------
__global__ __launch_bounds__(256) void select_kernel(const float4* __restrict__ x,
                                                     const int* __restrict__ frame_mask,
                                                     const int* __restrict__ graph_batch,
                                                     const int* __restrict__ med,
                                                     int* __restrict__ counter,
                                                     int* __restrict__ map,
                                                     int* __restrict__ sel_nodes,
                                                     float4* __restrict__ agg,
                                                     int n, int s_cap) {
    int i = blockIdx.x * 256 + threadIdx.x;
    if (i >= n) return;
    int slot = -1;
    int g = graph_batch[i];
    if (frame_mask[i] == med[g]) {
        int c = atomicAdd(counter, 1);
        if (c < s_cap) {
            slot = c;
            sel_nodes[c] = i;
            const float4* xs = x + (long)i * (D_IN / 4);
            float4* ap = agg + (long)c * (D_IN / 4);
#pragma unroll
            for (int k = 0; k < D_IN / 4; ++k) ap[k] = xs[k];
        }
    }
    map[i] = slot;
}

// ---------------------------------------------------------------------------
// 4) Edge scatter: only edges whose dst is a selected node do work (~4%).
//    Hardware f32 atomics (global_atomic_add_f32) into the compact agg rows.
// ---------------------------------------------------------------------------
__global__ __launch_bounds__(256) void edge_kernel(const int* __restrict__ src,
                                                   const int* __restrict__ dst,
                                                   const int* __restrict__ map,
                                                   const float* __restrict__ x,
                                                   float* __restrict__ agg,
                                                   int e_count) {
    int e = blockIdx.x * 256 + threadIdx.x;
    if (e >= e_count) return;
    int c = map[dst[e]];
    if (c < 0) return;
    const float* xr = x + (long)src[e] * D_IN;
    float* ar = agg + (long)c * D_IN;
#pragma unroll
    for (int k = 0; k < D_IN; ++k) {
        unsafeAtomicAdd(&ar[k], xr[k]);
    }
}

// ---------------------------------------------------------------------------
// 5) WMMA GEMM over selected rows: embbed = relu(agg @ W + b), fused with
//    per-graph masked-sum pooling via atomics. One wave per 16x16 tile,
//    K=64 via 16 chained V_WMMA_F32_16X16X4_F32.
//
//    f32 A 16x4 layout (ISA 7.12.2): lanes 0-15: M=lane, K={0,1} in v0/v1;
//    lanes 16-31: M=lane-16, K={2,3}.  B 4x16 mirrors it across lanes.
//    C/D: vgpr v, lanes 0-15 -> M=v, N=lane; lanes 16-31 -> M=v+8, N=lane-16.
// ---------------------------------------------------------------------------
__global__ __launch_bounds__(32) void gemm_pool_kernel(const float* __restrict__ agg,
                                                       const int* __restrict__ sel_nodes,
                                                       const int* __restrict__ counter,
                                                       int s_cap,
                                                       const float* __restrict__ W,
                                                       const float* __restrict__ bias,
                                                       const int* __restrict__ graph_batch,
                                                       float* __restrict__ num) {
    int s_total = *counter;
    if (s_total > s_cap) s_total = s_cap;
    const int tm = blockIdx.x;
    if (tm * 16 >= s_total) return;           // uniform early-exit, EXEC stays all-1s
    const int tn   = blockIdx.y;
    const int lane = threadIdx.x;             // wave32
    const int half = lane >> 4;               // 0 or 1
    const int l15  = lane & 15;

    int rowA = tm * 16 + l15;
    if (rowA > s_total - 1) rowA = s_total - 1;   // clamp pad rows (reads only)
    const float* aptr = agg + (long)rowA * D_IN + half * 2;

    const int col = tn * 16 + l15;
    const float* wptr = W + (half * 2) * H_DIM + col;

    v8f c = {0.f, 0.f, 0.f, 0.f, 0.f, 0.f, 0.f, 0.f};
#pragma unroll
    for (int j = 0; j < D_IN / 4; ++j) {
        v2f a, b;
        a.x = aptr[4 * j];
        a.y = aptr[4 * j + 1];
        b.x = wptr[(4 * j) * H_DIM];
        b.y = wptr[(4 * j + 1) * H_DIM];
        c = __builtin_amdgcn_wmma_f32_16x16x4_f32(false, a, false, b,
                                                  (short)0, c, false, false);
    }

    const float bv = bias[col];
#pragma unroll
    for (int v = 0; v < 8; ++v) {
        int rglob = tm * 16 + v + half * 8;
        if (rglob < s_total) {
            float val = c[v] + bv;
            val = val > 0.0f ? val : 0.0f;    // relu
            int node = sel_nodes[rglob];
            int g = graph_batch[node];
            unsafeAtomicAdd(&num[g * H_DIM + col], val);
        }
    }
}

// ---------------------------------------------------------------------------
// 6) Head: out[g,c] = (num[g,:]/den[g]) . Wc[:,c] + bc[c]   (1024x10, tiny)
// ---------------------------------------------------------------------------
__global__ __launch_bounds__(256) void head_kernel(const float* __restrict__ num,
                                                   const int* __restrict__ den,
                                                   const float* __restrict__ Wc,
                                                   const float* __restrict__ bc,
                                                   float* __restrict__ out, int total) {
    int t = blockIdx.x * 256 + threadIdx.x;
    if (t >= total) return;
    int g = t / C_CLS, cc = t % C_CLS;
    float inv = 1.0f / (float)den[g];
    const float* nr = num + (long)g * H_DIM;
    float acc = 0.0f;
#pragma unroll 8
    for (int h = 0; h < H_DIM; ++h) acc += nr[h] * Wc[h * C_CLS + cc];
    out[t] = acc * inv + bc[cc];
}

// ---------------------------------------------------------------------------
extern "C" void kernel_launch(void* const* d_in, const int* in_sizes, int n_in,
                              void* d_out, int out_size, void* d_ws, size_t ws_size,
                              hipStream_t stream) {
    const float* x          = (const float*)d_in[0];
    const int*   ei         = (const int*)d_in[1];
    const int*   frame_mask = (const int*)d_in[2];
    const int*   graph_batch= (const int*)d_in[3];
    const float* W          = (const float*)d_in[4];
    const float* bias       = (const float*)d_in[5];
    const float* Wc         = (const float*)d_in[6];
    const float* bcv        = (const float*)d_in[7];
    float*       out        = (float*)d_out;

    const int N = in_sizes[2];          // total nodes
    const int E = in_sizes[1] / 2;      // edges
    const int B = B_GRAPHS;

    // Workspace layout (16B aligned slices)
    char*  ws  = (char*)d_ws;
    size_t off = 0;
    auto alloc = [&](size_t bytes) -> char* {
        char* p = ws + off;
        off = (off + bytes + 15) & ~(size_t)15;
        return p;
    };
    int*   counter   = (int*)  alloc(4);
    int*   med       = (int*)  alloc((size_t)B * 4);
    int*   den       = (int*)  alloc((size_t)B * 4);
    int*   map       = (int*)  alloc((size_t)N * 4);
    float* num       = (float*)alloc((size_t)B * H_DIM * 4);
    size_t rem   = (ws_size > off) ? (ws_size - off - 32) : 0;
    long   s_cap = (long)(rem / (4 + D_IN * 4));   // sel_nodes + agg row per node
    if (s_cap > N) s_cap = N;
    if (s_cap < 1) s_cap = 1;
    int*   sel_nodes = (int*)  alloc((size_t)s_cap * 4);
    float* agg       = (float*)alloc((size_t)s_cap * D_IN * 4);

    const int* srcp = ei;
    const int* dstp = ei + E;

    median_kernel<<<B, 256, 0, stream>>>(frame_mask, med, den);
    zero_kernel<<<(B * H_DIM + 255) / 256, 256, 0, stream>>>(num, counter, B * H_DIM);
    select_kernel<<<(N + 255) / 256, 256, 0, stream>>>((const float4*)x, frame_mask,
                                                       graph_batch, med, counter, map,
                                                       sel_nodes, (float4*)agg, N, (int)s_cap);
    edge_kernel<<<(E + 255) / 256, 256, 0, stream>>>(srcp, dstp, map, x, agg, E);

    dim3 ggrid((unsigned)((s_cap + 15) / 16), H_DIM / 16);
    gemm_pool_kernel<<<ggrid, 32, 0, stream>>>(agg, sel_nodes, counter, (int)s_cap,
                                               W, bias, graph_batch, num);

    head_kernel<<<(B * C_CLS + 255) / 256, 256, 0, stream>>>(num, den, Wc, bcv, out,
                                                             B * C_CLS);
}